// Correlation_CrossEntropyLoss_88021059764894
// MI455X (gfx1250) — compile-verified
//
#include <hip/hip_runtime.h>
#include <hip/hip_bf16.h>
#include <stdint.h>

// Problem shape from the reference (compile-time for static LDS sizing).
#define C_DIM 32000
#define VEC4  (C_DIM / 4)   // 8000 float4 tiles per row
#define BLK   1024          // 32 waves per row-block
#define NWAVE (BLK / 32)
#define TPT   8             // float4 tiles per thread (8*1024 >= 8000)

__device__ __forceinline__ float wave_max(float v) {
#pragma unroll
    for (int o = 16; o > 0; o >>= 1) v = fmaxf(v, __shfl_xor(v, o, 32));
    return v;
}
__device__ __forceinline__ float wave_sum(float v) {
#pragma unroll
    for (int o = 16; o > 0; o >>= 1) v += __shfl_xor(v, o, 32);
    return v;
}

// One workgroup per batch row.
//   Phase 0: stream the 128 KB row HBM -> LDS exactly once (ASYNCcnt DMA path,
//            no VGPR landing slots, overlapped with the sibling block's compute).
//   Phase 1: ONE ds_load_b128 sweep into 32 VGPRs/thread; max from registers.
//   Phase 2: exp-sum from the same registers (zero extra LDS traffic).
//   Phase 3: three gathered columns from LDS + scalar loss terms.
__global__ __launch_bounds__(BLK)
void corr_ce_rows(const float* __restrict__ logits,
                  const int*   __restrict__ target,
                  const float* __restrict__ X1,
                  const int*   __restrict__ Y1,
                  const float* __restrict__ X2,
                  const int*   __restrict__ Y2,
                  const float* __restrict__ Tp,
                  float* __restrict__ per_row)  // [B][4]: loss, k, z, j
{
    __shared__ __align__(16) float row[C_DIM];   // 128 KB of the 320 KB/WGP LDS
    __shared__ float red[NWAVE];
    __shared__ float bcast;

    const int tid  = threadIdx.x;
    const int lane = tid & 31;
    const int wv   = tid >> 5;
    const int b    = blockIdx.x;

    const float*   src      = logits + (size_t)b * C_DIM;
    const uint32_t lds_base = (uint32_t)(uintptr_t)row;  // low 32 bits = LDS offset

    // ---- Phase 0: async DMA the row into LDS (CDNA5 GLOBAL_LOAD_ASYNC_TO_LDS_B128)
    for (int i = tid; i < VEC4; i += BLK) {
        uint32_t laddr = lds_base + (uint32_t)i * 16u;
        uint64_t gaddr = (uint64_t)(uintptr_t)(src + i * 4);
        asm volatile("global_load_async_to_lds_b128 %0, %1, off"
                     :: "v"(laddr), "v"(gaddr) : "memory");
    }
    asm volatile("s_wait_asynccnt 0x0" ::: "memory");
    __syncthreads();

    // ---- Phase 1: single LDS sweep into registers; padding is -INF (neutral
    //      for max, and exp2(-inf)=0 is neutral for the exp-sum).
    const float NEG = -3.402823466e+38f * 0.0f - 3.402823466e+38f; // -inf-ish guard
    float4 r[TPT];
#pragma unroll
    for (int k = 0; k < TPT; ++k) {
        const int i = tid + k * BLK;
        if (i < VEC4) {
            r[k] = reinterpret_cast<const float4*>(row)[i];  // ds_load_b128
        } else {
            r[k].x = r[k].y = r[k].z = r[k].w = NEG;
        }
    }

    float m = NEG;
#pragma unroll
    for (int k = 0; k < TPT; ++k)
        m = fmaxf(m, fmaxf(fmaxf(r[k].x, r[k].y), fmaxf(r[k].z, r[k].w)));
    m = wave_max(m);
    if (lane == 0) red[wv] = m;
    __syncthreads();
    if (tid < NWAVE) {
        float v = wave_max(red[tid]);
        if (tid == 0) bcast = v;
    }
    __syncthreads();
    const float M = bcast;

    // ---- Phase 2: sum exp(x - M) from registers; one v_exp_f32 per element.
    const float L2E = 1.44269504088896340736f;  // log2(e)
    const float mb  = M * L2E;                  // exp(x-M) = exp2(x*L2E - mb)
    float s = 0.0f;
#pragma unroll
    for (int k = 0; k < TPT; ++k) {
        s += __builtin_amdgcn_exp2f(__builtin_fmaf(r[k].x, L2E, -mb));
        s += __builtin_amdgcn_exp2f(__builtin_fmaf(r[k].y, L2E, -mb));
        s += __builtin_amdgcn_exp2f(__builtin_fmaf(r[k].z, L2E, -mb));
        s += __builtin_amdgcn_exp2f(__builtin_fmaf(r[k].w, L2E, -mb));
    }
    s = wave_sum(s);
    if (lane == 0) red[wv] = s;
    __syncthreads();
    if (tid < NWAVE) {
        float v = wave_sum(red[tid]);
        if (tid == 0) bcast = v;
    }
    __syncthreads();
    const float S = bcast;

    // ---- Phase 3: per-row terms from three gathered columns (thread 0)
    if (tid == 0) {
        const int   tgt  = target[b];
        const float xt   = row[tgt];
        const int   y1   = Y1[tgt];
        const int   y2   = Y2[tgt];
        const float x1   = X1[tgt];
        const float x2   = X2[tgt];
        const float invS = 1.0f / S;
        const float pt = __builtin_amdgcn_exp2f(__builtin_fmaf(xt,      L2E, -mb)) * invS;
        const float P1 = __builtin_amdgcn_exp2f(__builtin_fmaf(row[y1], L2E, -mb)) * invS;
        const float P2 = __builtin_amdgcn_exp2f(__builtin_fmaf(row[y2], L2E, -mb)) * invS;
        const float t    = Tp[0];
        const float corr = t * (x1 * P1 + x2 * P2);
        const bool  cond = pt > corr;
        const float LN2  = 0.69314718055994530942f;
        const float arg  = cond ? (pt - corr) : pt;        // -log(pt-corr) else -log(pt)
        const float loss = -LN2 * __builtin_amdgcn_logf(arg);
        const bool  nz   = (P1 != 0.0f) || (P2 != 0.0f);
        const bool  k    = cond && nz;
        const float z    = k ? pt / corr : 0.0f;
        float4 o;
        o.x = loss;
        o.y = k ? 1.0f : 0.0f;
        o.z = z;
        o.w = cond ? 0.0f : 1.0f;
        reinterpret_cast<float4*>(per_row)[b] = o;
    }
}

// Deterministic final reduction of [B][4] -> 4 outputs (loss scaled by 1/B).
__global__ __launch_bounds__(BLK)
void corr_ce_reduce(const float* __restrict__ per_row, float* __restrict__ out, int B)
{
    __shared__ float red[NWAVE];
    const int tid  = threadIdx.x;
    const int lane = tid & 31;
    const int wv   = tid >> 5;
    for (int c = 0; c < 4; ++c) {
        float s = 0.0f;
        for (int i = tid; i < B; i += BLK) s += per_row[4 * i + c];
        s = wave_sum(s);
        if (lane == 0) red[wv] = s;
        __syncthreads();
        if (tid < NWAVE) {
            float v = wave_sum(red[tid]);
            if (tid == 0) out[c] = (c == 0) ? v / (float)B : v;
        }
        __syncthreads();
    }
}

extern "C" void kernel_launch(void* const* d_in, const int* in_sizes, int n_in,
                              void* d_out, int out_size, void* d_ws, size_t ws_size,
                              hipStream_t stream) {
    // setup_inputs order: input, target, X1, Y1, X2, Y2, T
    const float* logits = (const float*)d_in[0];
    const int*   target = (const int*)  d_in[1];
    const float* X1     = (const float*)d_in[2];
    const int*   Y1     = (const int*)  d_in[3];
    const float* X2     = (const float*)d_in[4];
    const int*   Y2     = (const int*)  d_in[5];
    const float* T      = (const float*)d_in[6];
    const int    B      = in_sizes[1];          // 4096 (C is compile-time 32000)

    float* per_row = (float*)d_ws;              // B*4 floats = 64 KB scratch

    corr_ce_rows<<<B, BLK, 0, stream>>>(logits, target, X1, Y1, X2, Y2, T, per_row);
    corr_ce_reduce<<<1, BLK, 0, stream>>>(per_row, (float*)d_out, B);
}